// RelativeGlobalAttention_relative_index_pitch_onset_16346645528999
// MI455X (gfx1250) — compile-verified
//
#include <hip/hip_runtime.h>

typedef __attribute__((ext_vector_type(2))) float v2f;
typedef __attribute__((ext_vector_type(8))) float v8f;
typedef int async_v4i __attribute__((vector_size(16)));

#define B_  2
#define L_  256
#define D_  256
#define H_  4
#define DH_ 64

// ---------------------------------------------------------------------------
// CDNA5 async global->LDS copy (ASYNCcnt-tracked), builtin if present,
// inline-asm fallback otherwise. Param type per hipcc diagnostic:
//   (AS1 v4i*, AS3 v4i*, int offset, int cpol)
// ---------------------------------------------------------------------------
#if defined(__has_builtin)
#if __has_builtin(__builtin_amdgcn_global_load_async_to_lds_b128)
#define ASYNC_CP(g, l)                                                        \
  __builtin_amdgcn_global_load_async_to_lds_b128(                             \
      (__attribute__((address_space(1))) async_v4i*)(g),                      \
      (__attribute__((address_space(3))) async_v4i*)(l), 0, 0)
#define ASYNC_CP_DEFINED 1
#endif
#endif
#ifndef ASYNC_CP_DEFINED
#define ASYNC_CP(g, l)                                                        \
  asm volatile("global_load_async_to_lds_b128 %0, %1, off" ::                 \
               "v"((unsigned)(__SIZE_TYPE__)(                                 \
                   __attribute__((address_space(3))) async_v4i*)(l)),         \
               "v"((__attribute__((address_space(1))) async_v4i*)(g))         \
               : "memory")
#endif

#if defined(__has_builtin)
#if __has_builtin(__builtin_amdgcn_s_wait_asynccnt)
#define WAIT_ASYNC(n) __builtin_amdgcn_s_wait_asynccnt(n)
#define WAIT_ASYNC_DEFINED 1
#endif
#endif
#ifndef WAIT_ASYNC_DEFINED
#define WAIT_ASYNC(n) asm volatile("s_wait_asynccnt " #n ::: "memory")
#endif

// ---------------------------------------------------------------------------
// One wave computes a 16x16 fp32 tile with V_WMMA_F32_16X16X4_F32.
// Lane layout (ISA 7.12.2): A 16x4: lane&15 = M, k-pair = 2*(lane>>4)+{0,1}
//                           B 4x16: lane&15 = N, same k-pair split
//                           C/D:    M = r + 8*(lane>>4), N = lane&15
// ---------------------------------------------------------------------------

// C[m][n] = sum_k A[m*lda+k] * Bm[n*ldb+k]   ("NT": both k-contiguous)
__device__ __forceinline__ v8f tile_nt(const float* __restrict__ A, int lda,
                                       const float* __restrict__ Bm, int ldb,
                                       int K, v8f c) {
  const int lane = threadIdx.x & 31;
  const int lo = lane & 15, hi = lane >> 4;
#pragma unroll 4
  for (int kk = 0; kk < K; kk += 4) {
    const int k0 = kk + 2 * hi;
    v2f a = *(const v2f*)(A + lo * lda + k0);
    v2f b = *(const v2f*)(Bm + lo * ldb + k0);
    c = __builtin_amdgcn_wmma_f32_16x16x4_f32(false, a, false, b,
                                              (short)0, c, false, false);
  }
  return c;
}

// C[m][n] = sum_k A[m*lda+k] * Bm[k*ldb+n]   ("NN": B k-strided)
__device__ __forceinline__ v8f tile_nn(const float* __restrict__ A, int lda,
                                       const float* __restrict__ Bm, int ldb,
                                       int K, v8f c) {
  const int lane = threadIdx.x & 31;
  const int lo = lane & 15, hi = lane >> 4;
#pragma unroll 4
  for (int kk = 0; kk < K; kk += 4) {
    const int k0 = kk + 2 * hi;
    v2f a = *(const v2f*)(A + lo * lda + k0);
    v2f b;
    b.x = Bm[(size_t)k0 * ldb + lo];
    b.y = Bm[(size_t)(k0 + 1) * ldb + lo];
    c = __builtin_amdgcn_wmma_f32_16x16x4_f32(false, a, false, b,
                                              (short)0, c, false, false);
  }
  return c;
}

__device__ __forceinline__ void store_tile(float* __restrict__ C, int ldc, v8f c,
                                           const float* __restrict__ bias) {
  const int lane = threadIdx.x & 31;
  const int lo = lane & 15, hi = lane >> 4;
  const float bv = bias ? bias[lo] : 0.0f;
#pragma unroll
  for (int r = 0; r < 8; ++r)
    C[(size_t)(r + 8 * hi) * ldc + lo] = c[r] + bv;
}

// ---------------------------------------------------------------------------
// K1: y{q,k,v} = x @ W^T + b    (3 x (512,256)x(256,256))
// ---------------------------------------------------------------------------
__global__ void k_qkv(const float* __restrict__ q, const float* __restrict__ kx,
                      const float* __restrict__ vx,
                      const float* __restrict__ Wq, const float* __restrict__ bq,
                      const float* __restrict__ Wk, const float* __restrict__ bk,
                      const float* __restrict__ Wv, const float* __restrict__ bv,
                      float* __restrict__ yq, float* __restrict__ yk,
                      float* __restrict__ yv) {
  int t = blockIdx.x;                 // 3 * 32 * 16 = 1536 waves
  const int which = t >> 9; t &= 511;
  const int mt = t >> 4, nt = t & 15;
  const float* x  = which == 0 ? q  : which == 1 ? kx : vx;
  const float* W  = which == 0 ? Wq : which == 1 ? Wk : Wv;
  const float* bb = which == 0 ? bq : which == 1 ? bk : bv;
  float*       y  = which == 0 ? yq : which == 1 ? yk : yv;
  v8f c = {};
  c = tile_nt(x + (size_t)mt * 16 * D_, D_, W + (size_t)nt * 16 * D_, D_, D_, c);
  store_tile(y + (size_t)mt * 16 * D_ + nt * 16, D_, c, bb + nt * 16);
}

// ---------------------------------------------------------------------------
// K2: w_q[b,i,h,:] = qh[b,i,h*DH:..] @ {pr,dr}_w[h*DH:.., :]   (K=64)
// ---------------------------------------------------------------------------
__global__ void k_wq(const float* __restrict__ yq, const float* __restrict__ prw,
                     const float* __restrict__ drw,
                     float* __restrict__ wqp, float* __restrict__ wqd) {
  int t = blockIdx.x;                 // 2 * 4 * 32 * 16 = 4096 waves
  const int which = t >> 11; t &= 2047;
  const int h = t >> 9;      t &= 511;
  const int mt = t >> 4, nt = t & 15;
  const float* W = which ? drw : prw;
  float* outp    = which ? wqd : wqp;
  v8f c = {};
  c = tile_nn(yq + (size_t)mt * 16 * D_ + h * DH_, D_,
              W + (size_t)h * DH_ * D_ + nt * 16, D_, DH_, c);
  store_tile(outp + (size_t)mt * 16 * (H_ * D_) + h * D_ + nt * 16,
             H_ * D_, c, nullptr);
}

// ---------------------------------------------------------------------------
// K2b: csum[b,h,i] = qh . (pr_b + dr_b) per head
// ---------------------------------------------------------------------------
__global__ void k_csum(const float* __restrict__ yq, const float* __restrict__ prb,
                       const float* __restrict__ drb, float* __restrict__ csum) {
  const int tid = blockIdx.x * blockDim.x + threadIdx.x;   // 2048
  if (tid >= B_ * H_ * L_) return;
  const int i = tid & 255, h = (tid >> 8) & 3, b = tid >> 10;
  const float* qrow = yq + ((size_t)b * L_ + i) * D_ + h * DH_;
  float s = 0.0f;
  for (int d = 0; d < DH_; ++d) s += qrow[d] * (prb[h * DH_ + d] + drb[h * DH_ + d]);
  csum[tid] = s;                       // layout (b,h,i)
}

// ---------------------------------------------------------------------------
// K3: content logits[b,h,i,j] = qh . kh   (K=64)
// ---------------------------------------------------------------------------
__global__ void k_logits(const float* __restrict__ yq, const float* __restrict__ yk,
                         float* __restrict__ logits) {
  int t = blockIdx.x;                 // 8 * 16 * 16 = 2048 waves
  const int bh = t >> 8; t &= 255;
  const int mt = t >> 4, nt = t & 15;
  const int b = bh >> 2, h = bh & 3;
  v8f c = {};
  c = tile_nt(yq + ((size_t)b * L_ + mt * 16) * D_ + h * DH_, D_,
              yk + ((size_t)b * L_ + nt * 16) * D_ + h * DH_, D_, DH_, c);
  store_tile(logits + ((size_t)bh * L_ + mt * 16) * L_ + nt * 16, L_, c, nullptr);
}

// ---------------------------------------------------------------------------
// K4: HBM-bound stream: logits[b,h,i,j] += rel rows . w_q + csum
// Double-buffered async global->LDS pipeline (ASYNCcnt), 16-row chunks.
// One block per (b,i); lane owns an 8-float D-chunk, weights in registers.
// ---------------------------------------------------------------------------
#define CH_ 16                               // rows per chunk
#define CHF_ (CH_ * D_)                      // floats per tensor per chunk

__global__ void __launch_bounds__(256) k_bias(
    const float* __restrict__ pr, const float* __restrict__ dr,
    const float* __restrict__ wqp, const float* __restrict__ wqd,
    const float* __restrict__ csum, float* __restrict__ logits) {
  const int bi = blockIdx.x;           // 512 = b*L + i
  const int b = bi >> 8, i = bi & 255;
  const int tid = threadIdx.x;
  const int lane = tid & 31;
  const int wave = tid >> 5;           // 8 waves
  const int d0 = lane * 8;

  __shared__ float lds[2 * 2 * CHF_];  // [buf][pitch|dur][16 rows][256] = 64KB

  // register-resident per-(b,i) weight chunks + bias constants
  float wp[H_][8], wd[H_][8];
  const float* wpb = wqp + (size_t)bi * (H_ * D_);
  const float* wdb = wqd + (size_t)bi * (H_ * D_);
#pragma unroll
  for (int h = 0; h < H_; ++h)
#pragma unroll
    for (int t = 0; t < 8; ++t) {
      wp[h][t] = wpb[h * D_ + d0 + t];
      wd[h][t] = wdb[h * D_ + d0 + t];
    }
  float cs[H_];
#pragma unroll
  for (int h = 0; h < H_; ++h) cs[h] = csum[((size_t)b * H_ + h) * L_ + i];

  const float* pbase = pr + (size_t)bi * L_ * D_;   // contiguous 256KB slab
  const float* dbase = dr + (size_t)bi * L_ * D_;

  // issue one chunk: 16KB/tensor = 4 block-wide b128 issues per tensor
  auto issue_chunk = [&](int c) {
    float* lp = lds + (c & 1) * (2 * CHF_);
    float* ld = lp + CHF_;
    const float* gp = pbase + (size_t)c * CHF_;
    const float* gd = dbase + (size_t)c * CHF_;
#pragma unroll
    for (int it = 0; it < 4; ++it) {
      const int off = it * 1024 + tid * 4;           // floats, 16B per lane
      ASYNC_CP(gp + off, lp + off);
      ASYNC_CP(gd + off, ld + off);
    }
  };

  const int NCHUNK = L_ / CH_;         // 16
  issue_chunk(0);
  issue_chunk(1);

  for (int c = 0; c < NCHUNK; ++c) {
    if (c + 1 < NCHUNK) { WAIT_ASYNC(8); }   // chunk c done, c+1 in flight
    else                { WAIT_ASYNC(0); }
    __syncthreads();                          // all waves' copies landed

    const float* lp = lds + (c & 1) * (2 * CHF_);
    const float* ld = lp + CHF_;

#pragma unroll
    for (int rr = 0; rr < 2; ++rr) {          // 8 waves x 2 rows = 16 rows
      const int r = wave * 2 + rr;
      const int j = c * CH_ + r;
      const float4 p0 = ((const float4*)(lp + r * D_ + d0))[0];
      const float4 p1 = ((const float4*)(lp + r * D_ + d0))[1];
      const float4 q0 = ((const float4*)(ld + r * D_ + d0))[0];
      const float4 q1 = ((const float4*)(ld + r * D_ + d0))[1];
      float s[H_];
#pragma unroll
      for (int h = 0; h < H_; ++h) {
        s[h] = p0.x * wp[h][0] + p0.y * wp[h][1] + p0.z * wp[h][2] + p0.w * wp[h][3]
             + p1.x * wp[h][4] + p1.y * wp[h][5] + p1.z * wp[h][6] + p1.w * wp[h][7]
             + q0.x * wd[h][0] + q0.y * wd[h][1] + q0.z * wd[h][2] + q0.w * wd[h][3]
             + q1.x * wd[h][4] + q1.y * wd[h][5] + q1.z * wd[h][6] + q1.w * wd[h][7];
      }
#pragma unroll
      for (int off = 16; off > 0; off >>= 1)
#pragma unroll
        for (int h = 0; h < H_; ++h) s[h] += __shfl_xor(s[h], off, 32);
      if (lane == 0) {
#pragma unroll
        for (int h = 0; h < H_; ++h)
          logits[(((size_t)b * H_ + h) * L_ + i) * L_ + j] += s[h] + cs[h];
      }
    }

    __syncthreads();                          // buffer free before re-issue
    if (c + 2 < NCHUNK) issue_chunk(c + 2);
  }
}

// ---------------------------------------------------------------------------
// K5: scale + mask + softmax -> attn (written straight into d_out region)
// ---------------------------------------------------------------------------
__global__ void __launch_bounds__(256) k_softmax(
    const float* __restrict__ logits, const float* __restrict__ amask,
    const unsigned char* __restrict__ kpm, float* __restrict__ attn) {
  const int row = blockIdx.x;          // 2048 = (b*H+h)*L + i
  const int i = row & 255;
  const int b = row >> 10;
  const int j = threadIdx.x;
  __shared__ float red[256];

  const float lg = logits[(size_t)row * L_ + j] * 0.125f;  // 1/sqrt(64)
  const bool m = (amask[(size_t)i * L_ + j] != 0.0f) || (kpm[b * L_ + j] != 0);
  const float val = m ? -9.0e15f : lg;

  red[j] = val; __syncthreads();
  for (int s = 128; s > 0; s >>= 1) {
    if (j < s) red[j] = fmaxf(red[j], red[j + s]);
    __syncthreads();
  }
  const float mx = red[0]; __syncthreads();
  const float e = __expf(val - mx);
  red[j] = e; __syncthreads();
  for (int s = 128; s > 0; s >>= 1) {
    if (j < s) red[j] += red[j + s];
    __syncthreads();
  }
  attn[(size_t)row * L_ + j] = e * (1.0f / red[0]);
}

// ---------------------------------------------------------------------------
// K6: out_heads[b,i,h*DH+n] = sum_j attn[b,h,i,j] * vh[b,j,h*DH+n]   (K=256)
// ---------------------------------------------------------------------------
__global__ void k_av(const float* __restrict__ attn, const float* __restrict__ yv,
                     float* __restrict__ oh) {
  int t = blockIdx.x;                  // 8 * 16 * 4 = 512 waves
  const int bh = t >> 6; t &= 63;
  const int mt = t >> 2, nt = t & 3;
  const int b = bh >> 2, h = bh & 3;
  v8f c = {};
  c = tile_nn(attn + ((size_t)bh * L_ + mt * 16) * L_, L_,
              yv + (size_t)b * L_ * D_ + h * DH_ + nt * 16, D_, L_, c);
  store_tile(oh + ((size_t)b * L_ + mt * 16) * D_ + h * DH_ + nt * 16,
             D_, c, nullptr);
}

// ---------------------------------------------------------------------------
// K7: out = out_heads @ fc_w^T + fc_b
// ---------------------------------------------------------------------------
__global__ void k_fc(const float* __restrict__ oh, const float* __restrict__ fw,
                     const float* __restrict__ fb, float* __restrict__ outp) {
  int t = blockIdx.x;                  // 32 * 16 = 512 waves
  const int mt = t >> 4, nt = t & 15;
  v8f c = {};
  c = tile_nt(oh + (size_t)mt * 16 * D_, D_, fw + (size_t)nt * 16 * D_, D_, D_, c);
  store_tile(outp + (size_t)mt * 16 * D_ + nt * 16, D_, c, fb + nt * 16);
}

// ---------------------------------------------------------------------------
extern "C" void kernel_launch(void* const* d_in, const int* in_sizes, int n_in,
                              void* d_out, int out_size, void* d_ws, size_t ws_size,
                              hipStream_t stream) {
  const float* q     = (const float*)d_in[0];
  const float* kx    = (const float*)d_in[1];
  const float* vx    = (const float*)d_in[2];
  const float* amask = (const float*)d_in[3];
  const unsigned char* kpm = (const unsigned char*)d_in[4];
  const float* pr    = (const float*)d_in[5];
  const float* dr    = (const float*)d_in[6];
  const float* Wq_w = (const float*)d_in[7];
  const float* Wq_b = (const float*)d_in[8];
  const float* Wk_w = (const float*)d_in[9];
  const float* Wk_b = (const float*)d_in[10];
  const float* Wv_w = (const float*)d_in[11];
  const float* Wv_b = (const float*)d_in[12];
  const float* fc_w = (const float*)d_in[13];
  const float* fc_b = (const float*)d_in[14];
  const float* pr_w = (const float*)d_in[15];
  const float* pr_b = (const float*)d_in[16];
  const float* dr_w = (const float*)d_in[17];
  const float* dr_b = (const float*)d_in[18];

  float* outp = (float*)d_out;                       // (B,L,D) = 131072
  float* attn = outp + (size_t)B_ * L_ * D_;         // (B,H,L,L) = 524288

  float* ws    = (float*)d_ws;
  float* yq    = ws;                    // 131072
  float* yk    = yq + 131072;           // 131072
  float* yv    = yk + 131072;           // 131072
  float* wqp   = yv + 131072;           // 524288
  float* wqd   = wqp + 524288;          // 524288
  float* csum  = wqd + 524288;          // 2048
  float* logit = csum + 2048;           // 524288
  float* oh    = logit + 524288;        // 131072   (total ~8.4 MB)

  k_qkv    <<<1536, 32, 0, stream>>>(q, kx, vx, Wq_w, Wq_b, Wk_w, Wk_b,
                                     Wv_w, Wv_b, yq, yk, yv);
  k_wq     <<<4096, 32, 0, stream>>>(yq, pr_w, dr_w, wqp, wqd);
  k_csum   <<<8, 256, 0, stream>>>(yq, pr_b, dr_b, csum);
  k_logits <<<2048, 32, 0, stream>>>(yq, yk, logit);
  k_bias   <<<512, 256, 0, stream>>>(pr, dr, wqp, wqd, csum, logit);
  k_softmax<<<2048, 256, 0, stream>>>(logit, amask, kpm, attn);
  k_av     <<<512, 32, 0, stream>>>(attn, yv, oh);
  k_fc     <<<512, 32, 0, stream>>>(oh, fc_w, fc_b, outp);
}